// OrthogonalMixing_90494960926927
// MI455X (gfx1250) — compile-verified
//
#include <hip/hip_runtime.h>

// ---------------------------------------------------------------------------
// Orthogonal polar factor of H (4096x4096 fp32) via Newton-Schulz iterations.
// All GEMMs on the CDNA5 fp32 matrix pipe (v_wmma_f32_16x16x4_f32), with
// double-buffered LDS staging and ping-pong fragment registers.
// ---------------------------------------------------------------------------

typedef float v2f __attribute__((ext_vector_type(2)));
typedef float v4f __attribute__((ext_vector_type(4)));
typedef float v8f __attribute__((ext_vector_type(8)));

#define NMAT 4096
#define BM 128
#define BN 128
#define BK 32
#define LDA 36          // A tile leading dim (k): 144B rows -> 16B aligned, conflict-free b64 frag reads
#define PAIR_STRIDE 288 // B pair-row stride: 256 + 32 pad so lane-half 1 lands on banks 32..63

// C = alpha * op(A) * B + beta * Dadd     (op = transpose if TA)
// A,B,Dadd,C are NMAT x NMAT row-major fp32.
template <bool TA>
__global__ __launch_bounds__(256) void gemm_ns(const float* __restrict__ A,
                                               const float* __restrict__ B,
                                               const float* __restrict__ Dadd,
                                               float* __restrict__ C,
                                               float alpha, float beta) {
  // Double-buffered tiles.
  __shared__ float As[2][BM * LDA];                // As[m*LDA + k]
  __shared__ float Bs[2][(BK / 2) * PAIR_STRIDE];  // Bs[(k>>1)*PS + n*2 + (k&1)]

  const int tid  = threadIdx.x;
  const int lane = tid & 31;
  const int wave = tid >> 5;          // 0..7
  const int wm0  = (wave >> 2) * 64;  // wave M origin
  const int wn0  = (wave & 3) * 32;   // wave N origin
  const int l15  = lane & 15;
  const int lh   = lane >> 4;         // lane half: selects K pair (frags) / M half (C)

  const int bm = blockIdx.x * BM;
  const int bn = blockIdx.y * BN;

  v8f acc[4][2];
#pragma unroll
  for (int i = 0; i < 4; ++i)
#pragma unroll
    for (int j = 0; j < 2; ++j)
#pragma unroll
      for (int r = 0; r < 8; ++r) acc[i][j][r] = 0.0f;

  // Staging registers (global -> reg -> LDS pipeline).
  v4f aReg[4];
  v4f bReg[4];

  // Global -> registers for K-tile at k0 (no LDS traffic; loads stay in flight).
  auto loadRegs = [&](int k0) {
#pragma unroll
    for (int it = 0; it < 4; ++it) {
      const int s = tid + it * 256;  // 0..1023 staging slots
      if (!TA) {
        const int r = s >> 3;        // row 0..127
        const int c = (s & 7) * 4;   // k   0..28
        aReg[it] = *reinterpret_cast<const v4f*>(A + (size_t)(bm + r) * NMAT + k0 + c);
      } else {
        // op(A)[m][k] = A[k][m]: thread owns 4 consecutive k of one column m.
        const int m  = s & 127;
        const int kq = (s >> 7) * 4;
        v4f v;
        v.x = A[(size_t)(k0 + kq + 0) * NMAT + bm + m];
        v.y = A[(size_t)(k0 + kq + 1) * NMAT + bm + m];
        v.z = A[(size_t)(k0 + kq + 2) * NMAT + bm + m];
        v.w = A[(size_t)(k0 + kq + 3) * NMAT + bm + m];
        aReg[it] = v;
      }
      // B: thread owns 4 consecutive k of one column n (coalesced per k-row).
      const int n  = s & 127;
      const int kq = (s >> 7) * 4;
      v4f v;
      v.x = B[(size_t)(k0 + kq + 0) * NMAT + bn + n];
      v.y = B[(size_t)(k0 + kq + 1) * NMAT + bn + n];
      v.z = B[(size_t)(k0 + kq + 2) * NMAT + bn + n];
      v.w = B[(size_t)(k0 + kq + 3) * NMAT + bn + n];
      bReg[it] = v;
    }
  };

  // Registers -> LDS buffer `buf`.
  auto storeRegs = [&](int buf) {
#pragma unroll
    for (int it = 0; it < 4; ++it) {
      const int s = tid + it * 256;
      if (!TA) {
        const int r = s >> 3;
        const int c = (s & 7) * 4;
        *reinterpret_cast<v4f*>(&As[buf][r * LDA + c]) = aReg[it];
      } else {
        const int m  = s & 127;
        const int kq = (s >> 7) * 4;
        *reinterpret_cast<v4f*>(&As[buf][m * LDA + kq]) = aReg[it];
      }
      // B pair-interleaved: two contiguous v2f stores, lanes span 64 banks.
      const int n  = s & 127;
      const int kq = (s >> 7) * 4;
      v2f lo; lo.x = bReg[it].x; lo.y = bReg[it].y;
      v2f hi; hi.x = bReg[it].z; hi.y = bReg[it].w;
      *reinterpret_cast<v2f*>(&Bs[buf][((kq >> 1) + 0) * PAIR_STRIDE + n * 2]) = lo;
      *reinterpret_cast<v2f*>(&Bs[buf][((kq >> 1) + 1) * PAIR_STRIDE + n * 2]) = hi;
    }
  };

  // Fragment loads (ISA 7.12.2 layouts): single b64 per fragment.
  auto loadFragsA = [&](v2f af[4], int buf, int kk) {
#pragma unroll
    for (int i = 0; i < 4; ++i) {
      const int row = wm0 + i * 16 + l15;
      af[i] = *reinterpret_cast<const v2f*>(&As[buf][row * LDA + kk + lh * 2]);
    }
  };
  auto loadFragsB = [&](v2f bf[2], int buf, int kk) {
#pragma unroll
    for (int j = 0; j < 2; ++j) {
      const int col = wn0 + j * 16 + l15;
      bf[j] = *reinterpret_cast<const v2f*>(&Bs[buf][((kk >> 1) + lh) * PAIR_STRIDE + col * 2]);
    }
  };
  auto mmaStep = [&](v2f af[4], v2f bf[2]) {
#pragma unroll
    for (int i = 0; i < 4; ++i)
#pragma unroll
      for (int j = 0; j < 2; ++j)
        acc[i][j] = __builtin_amdgcn_wmma_f32_16x16x4_f32(
            false, af[i], false, bf[j], (short)0, acc[i][j], false, false);
  };

  // Prologue: stage first K-tile.
  loadRegs(0);
  storeRegs(0);
  __syncthreads();

  for (int k0 = 0; k0 < NMAT; k0 += BK) {
    const int buf = (k0 / BK) & 1;
    const bool has_next = (k0 + BK) < NMAT;
    if (has_next) loadRegs(k0 + BK);  // next tile's global loads in flight under the MMAs

    // Ping-pong fragment registers: issue ds_loads one k-step ahead.
    v2f a0[4], b0[2], a1[4], b1[2];
    loadFragsA(a0, buf, 0);
    loadFragsB(b0, buf, 0);
#pragma unroll
    for (int kk = 0; kk < BK; kk += 8) {
      loadFragsA(a1, buf, kk + 4);
      loadFragsB(b1, buf, kk + 4);
      mmaStep(a0, b0);
      if (kk + 8 < BK) {
        loadFragsA(a0, buf, kk + 8);
        loadFragsB(b0, buf, kk + 8);
      }
      mmaStep(a1, b1);
    }

    if (has_next) storeRegs(buf ^ 1);  // drain global loads into the other buffer
    __syncthreads();
  }

  // Epilogue: C = alpha*acc + beta*Dadd.
  // C/D layout: VGPR r, lane L -> M = r + 8*(L>>4), N = L&15.
#pragma unroll
  for (int i = 0; i < 4; ++i) {
#pragma unroll
    for (int j = 0; j < 2; ++j) {
      const int col = bn + wn0 + j * 16 + l15;
#pragma unroll
      for (int r = 0; r < 8; ++r) {
        const int row = bm + wm0 + i * 16 + r + lh * 8;
        float v = alpha * acc[i][j][r];
        if (Dadd) v += beta * Dadd[(size_t)row * NMAT + col];
        C[(size_t)row * NMAT + col] = v;
      }
    }
  }
}

// ---- deterministic Frobenius-norm reduction (no float atomics) ----
__global__ __launch_bounds__(256) void frob_partial(const float* __restrict__ X,
                                                    float* __restrict__ part) {
  __shared__ float sdata[256];
  const size_t i0 = (size_t)blockIdx.x * (256 * 64) + threadIdx.x;
  float s = 0.0f;
#pragma unroll 4
  for (int k = 0; k < 64; ++k) {
    const float v = X[i0 + (size_t)k * 256];
    s += v * v;
  }
  sdata[threadIdx.x] = s;
  __syncthreads();
  for (int off = 128; off > 0; off >>= 1) {
    if (threadIdx.x < (unsigned)off) sdata[threadIdx.x] += sdata[threadIdx.x + off];
    __syncthreads();
  }
  if (threadIdx.x == 0) part[blockIdx.x] = sdata[0];
}

__global__ __launch_bounds__(1024) void frob_finalize(const float* __restrict__ part,
                                                      float* __restrict__ scal) {
  __shared__ float sdata[1024];
  sdata[threadIdx.x] = part[threadIdx.x];
  __syncthreads();
  for (int off = 512; off > 0; off >>= 1) {
    if (threadIdx.x < (unsigned)off) sdata[threadIdx.x] += sdata[threadIdx.x + off];
    __syncthreads();
  }
  if (threadIdx.x == 0) scal[0] = rsqrtf(sdata[0]);  // 1 / ||H||_F
}

__global__ __launch_bounds__(256) void scale_kernel(const float* __restrict__ H,
                                                    const float* __restrict__ scal,
                                                    float* __restrict__ X) {
  const size_t i = (size_t)blockIdx.x * blockDim.x + threadIdx.x;
  X[i] = H[i] * scal[0];
}

extern "C" void kernel_launch(void* const* d_in, const int* in_sizes, int n_in,
                              void* d_out, int out_size, void* d_ws, size_t ws_size,
                              hipStream_t stream) {
  (void)in_sizes; (void)n_in; (void)out_size; (void)ws_size;
  const float* H = (const float*)d_in[0];
  float* out = (float*)d_out;

  char* ws = (char*)d_ws;
  const size_t MATB = (size_t)NMAT * NMAT * sizeof(float);  // 64 MB
  float* X = (float*)(ws);
  float* P = (float*)(ws + MATB);
  float* Q = (float*)(ws + 2 * MATB);
  float* part = (float*)(ws + 3 * MATB);
  float* scal = part + 1024;

  // X0 = H / ||H||_F  (sigma_max <= 1, Newton-Schulz basin)
  frob_partial<<<1024, 256, 0, stream>>>(H, part);
  frob_finalize<<<1, 1024, 0, stream>>>(part, scal);
  scale_kernel<<<(NMAT * NMAT) / 256, 256, 0, stream>>>(H, scal, X);

  dim3 grid(NMAT / BM, NMAT / BN);
  dim3 blk(256);

  // Phase 1: aggressive quintic iterations X <- aX + bX(XtX) + cX(XtX)^2
  const float qa = 3.4445f, qb = -4.7750f, qc = 2.0315f;
  for (int it = 0; it < 10; ++it) {
    gemm_ns<true ><<<grid, blk, 0, stream>>>(X, X, nullptr, P, 1.0f, 0.0f);  // P = XtX
    gemm_ns<false><<<grid, blk, 0, stream>>>(P, P, P, Q, qc, qb);            // Q = c*P*P + b*P
    gemm_ns<false><<<grid, blk, 0, stream>>>(X, Q, X, P, 1.0f, qa);          // Pnew = X*Q + a*X
    float* t = X; X = P; P = t;
  }

  // Phase 2: cubic Newton-Schulz polish X <- 1.5X - 0.5 X (XtX); last writes d_out
  for (int it = 0; it < 6; ++it) {
    gemm_ns<true ><<<grid, blk, 0, stream>>>(X, X, nullptr, P, 1.0f, 0.0f);  // P = XtX
    float* dst = (it == 5) ? out : Q;
    gemm_ns<false><<<grid, blk, 0, stream>>>(X, P, X, dst, -0.5f, 1.5f);     // dst = 1.5X - 0.5*X*P
    if (it != 5) { float* t = X; X = Q; Q = t; }
  }
}